// CTCLoss_66357244723677
// MI455X (gfx1250) — compile-verified
//
#include <hip/hip_runtime.h>
#include <stdint.h>

#ifndef __has_builtin
#define __has_builtin(x) 0
#endif

#define N_DIM 128
#define T_DIM 1024
#define C_DIM 256
#define L_DIM 128
#define S_DIM 257          // 2*L+1
#define NEGF (-1e30f)

typedef float v2f __attribute__((ext_vector_type(2)));
typedef float v8f __attribute__((ext_vector_type(8)));

__device__ __forceinline__ float wave_max32(float x) {
  #pragma unroll
  for (int off = 16; off > 0; off >>= 1) x = fmaxf(x, __shfl_xor(x, off, 32));
  return x;
}
__device__ __forceinline__ float wave_sum32(float x) {
  #pragma unroll
  for (int off = 16; off > 0; off >>= 1) x += __shfl_xor(x, off, 32);
  return x;
}

// ---------------------------------------------------------------------------
// Kernel 1: logZ[n,t] = max_c preds[n,t,c] + log(sum_c exp(preds - max))
// One wave per (n,t) row; 8 rows per 256-thread block. Bandwidth bound.
// ---------------------------------------------------------------------------
__global__ __launch_bounds__(256) void ctc_logz(const float* __restrict__ preds,
                                                float* __restrict__ logZ) {
  const int wid  = threadIdx.x >> 5;
  const int lane = threadIdx.x & 31;
  const long long row = (long long)blockIdx.x * 8 + wid;   // 0 .. N*T-1
  const float* p = preds + row * C_DIM;

  float4 x0 = ((const float4*)p)[lane];        // c = 4*lane .. 4*lane+3
  float4 x1 = ((const float4*)p)[lane + 32];   // c = 128 + 4*lane ..
  float m = fmaxf(fmaxf(fmaxf(x0.x, x0.y), fmaxf(x0.z, x0.w)),
                  fmaxf(fmaxf(x1.x, x1.y), fmaxf(x1.z, x1.w)));
  m = wave_max32(m);
  float s = __expf(x0.x - m) + __expf(x0.y - m) + __expf(x0.z - m) + __expf(x0.w - m)
          + __expf(x1.x - m) + __expf(x1.y - m) + __expf(x1.z - m) + __expf(x1.w - m);
  s = wave_sum32(s);
  if (lane == 0) logZ[row] = m + __logf(s);
}

// ---------------------------------------------------------------------------
// Kernel 2: forward scan. One block (9 waves, 288 threads) per sample.
// Thread s (s<=256) owns alpha state s. Preds row (1KB) is double-buffered
// into LDS with CDNA5 async global->LDS loads issued by wave 0 two steps
// ahead; per-step critical path is just the banded logaddexp update.
// ---------------------------------------------------------------------------
__global__ __launch_bounds__(288) void ctc_scan(const float* __restrict__ preds,
                                                const int*   __restrict__ targets,
                                                const float* __restrict__ logZ,
                                                float*       __restrict__ losses) {
  __shared__ __align__(16) float sRow[2][C_DIM];
  __shared__ float sAlpha[2][S_DIM];

  const int n = blockIdx.x;
  const int s = threadIdx.x;
  const float* P     = preds + (size_t)n * T_DIM * C_DIM;
  const int*   tg    = targets + n * L_DIM;
  const float* lzrow = logZ + (size_t)n * T_DIM;

  // Per-state extended label class and skip-transition flag.
  int  cls  = 0;
  bool skip = false;
  if (s <= 256) {
    if (s & 1) {
      cls = tg[(s - 1) >> 1];                       // labels in [1,C)
      if (s >= 3) skip = (cls != tg[(s - 3) >> 1]); // ext[s] != ext[s-2]
    }
  }

  const uint32_t lane = threadIdx.x & 31;

  // Prologue: async-prefetch rows 0 and 1 (2 b128 instructions per row,
  // each lane moves 16B; ASYNCcnt += 2 per row, completes in order).
  if (threadIdx.x < 32) {
    uint64_t g0 = (uint64_t)(uintptr_t)P + lane * 16u;
    uint32_t l0 = (uint32_t)(uintptr_t)(&sRow[0][0]) + lane * 16u;
    asm volatile("global_load_async_to_lds_b128 %0, %1, off\n\t"
                 "global_load_async_to_lds_b128 %2, %3, off"
                 :: "v"(l0), "v"(g0),
                    "v"((uint32_t)(l0 + 512u)), "v"((uint64_t)(g0 + 512u))
                 : "memory");
    uint64_t g1 = (uint64_t)(uintptr_t)(P + C_DIM) + lane * 16u;
    uint32_t l1 = (uint32_t)(uintptr_t)(&sRow[1][0]) + lane * 16u;
    asm volatile("global_load_async_to_lds_b128 %0, %1, off\n\t"
                 "global_load_async_to_lds_b128 %2, %3, off"
                 :: "v"(l1), "v"(g1),
                    "v"((uint32_t)(l1 + 512u)), "v"((uint64_t)(g1 + 512u))
                 : "memory");
  }

  float lz_cur = lzrow[0];

  for (int t = 0; t < T_DIM; ++t) {
    float lz_nxt = (t + 1 < T_DIM) ? lzrow[t + 1] : 0.0f;   // prefetch normalizer

    // Row t complete once ASYNCcnt <= 2 (only row t+1's two ops may remain).
    if (threadIdx.x < 32) asm volatile("s_wait_asynccnt 0x2" ::: "memory");
    __syncthreads();   // publish LDS row t to all waves; alpha[t-1] visible

    const float* row = sRow[t & 1];
    const int wr = t & 1, rd = wr ^ 1;

    if (s <= 256) {
      float lp = row[cls] - lz_cur;   // fused log_softmax gather
      float na;
      if (t == 0) {
        na = (s < 2) ? lp : NEGF;
      } else {
        float a0 = sAlpha[rd][s];
        float a1 = (s >= 1) ? sAlpha[rd][s - 1] : NEGF;
        float a2 = skip     ? sAlpha[rd][s - 2] : NEGF;
        float m  = fmaxf(a0, fmaxf(a1, a2));
        float z  = __expf(a0 - m) + __expf(a1 - m) + __expf(a2 - m);
        na = m + __logf(z) + lp;
      }
      sAlpha[wr][s] = na;
    }
    __syncthreads();   // alpha[t] published; all waves done reading buf (t&1)

    // Buffer (t&1) is free: prefetch row t+2 into it.
    if (threadIdx.x < 32 && (t + 2) < T_DIM) {
      uint64_t g = (uint64_t)(uintptr_t)(P + (size_t)(t + 2) * C_DIM) + lane * 16u;
      uint32_t l = (uint32_t)(uintptr_t)(&sRow[t & 1][0]) + lane * 16u;
      asm volatile("global_load_async_to_lds_b128 %0, %1, off\n\t"
                   "global_load_async_to_lds_b128 %2, %3, off"
                   :: "v"(l), "v"(g),
                      "v"((uint32_t)(l + 512u)), "v"((uint64_t)(g + 512u))
                   : "memory");
    }
    lz_cur = lz_nxt;
  }

  if (threadIdx.x == 0) {
    int tl = 0;
    for (int i = 0; i < L_DIM; ++i) tl += (tg[i] != 0) ? 1 : 0;
    const int fin = (T_DIM - 1) & 1;
    float a = sAlpha[fin][2 * tl];
    float b = sAlpha[fin][2 * tl - 1];
    float m = fmaxf(a, b);
    float f = m + __logf(__expf(a - m) + __expf(b - m));
    float loss = (f < -1e29f) ? 0.0f : -f;             // zero_infinity
    int tld = (tl > 1) ? tl : 1;
    losses[n] = loss / (float)tld;                     // per-sample /tl
  }
}

// ---------------------------------------------------------------------------
// Kernel 3: mean over 128 per-sample losses with V_WMMA_F32_16X16X4_F32.
// D = A(16x4 of values) x B(4x16 of ones) + C  => row-sums; two WMMAs
// accumulate all 128 values, then 8 D-register adds + one xor-16 shuffle.
// ---------------------------------------------------------------------------
__global__ __launch_bounds__(32) void ctc_reduce(const float* __restrict__ losses,
                                                 float* __restrict__ out) {
  const int lane = threadIdx.x;
  const int mrow = lane & 15;
  const int hi   = lane >> 4;
  const int base = mrow * 4 + hi * 2;   // A[m][k] layout: lanes 0-15 k={0,1}; 16-31 k={2,3}
  float tot;
#if __has_builtin(__builtin_amdgcn_wmma_f32_16x16x4_f32)
  v2f a0; a0.x = losses[base];      a0.y = losses[base + 1];
  v2f a1; a1.x = losses[64 + base]; a1.y = losses[64 + base + 1];
  v2f b;  b.x = 1.0f; b.y = 1.0f;   // B = ones(4x16)
  v8f c = {};
  c = __builtin_amdgcn_wmma_f32_16x16x4_f32(false, a0, false, b, (short)0, c, false, false);
  c = __builtin_amdgcn_wmma_f32_16x16x4_f32(false, a1, false, b, (short)0, c, false, false);
  float part = c[0] + c[1] + c[2] + c[3] + c[4] + c[5] + c[6] + c[7];
  tot = part + __shfl_xor(part, 16, 32);   // lanes<16 hold rows 0-7, >=16 rows 8-15
#else
  float p = losses[lane] + losses[lane + 32] + losses[lane + 64] + losses[lane + 96];
  #pragma unroll
  for (int off = 16; off > 0; off >>= 1) p += __shfl_xor(p, off, 32);
  tot = p;
#endif
  if (lane == 0) out[0] = tot / (float)N_DIM;
}

// ---------------------------------------------------------------------------
extern "C" void kernel_launch(void* const* d_in, const int* in_sizes, int n_in,
                              void* d_out, int out_size, void* d_ws, size_t ws_size,
                              hipStream_t stream) {
  (void)in_sizes; (void)n_in; (void)out_size; (void)ws_size;
  const float* preds   = (const float*)d_in[0];
  const int*   targets = (const int*)d_in[1];
  float* ws     = (float*)d_ws;
  float* logZ   = ws;                              // N*T floats (512 KB)
  float* losses = ws + (size_t)N_DIM * T_DIM;      // 128 floats
  float* out    = (float*)d_out;

  ctc_logz  <<<(N_DIM * T_DIM) / 8, 256, 0, stream>>>(preds, logZ);
  ctc_scan  <<<N_DIM, 288, 0, stream>>>(preds, targets, logZ, losses);
  ctc_reduce<<<1, 32, 0, stream>>>(losses, out);
}